// ModelNew_60713657697089
// MI455X (gfx1250) — compile-verified
//
#include <hip/hip_runtime.h>
#include <math.h>

// MLA paged-attention decode, fused flash-decode (split-K) for gfx1250.
//
// Roofline: memory bound. Fusing score->softmax->PV per 128-token page makes
// the second read of each kv_nope page an L2 hit (192MB L2, ~128MB of live
// pages across 512 concurrent WGs), so HBM traffic ~= ONE pass over the KV
// cache (~350MB avg -> ~15us floor at 23.3 TB/s) vs 2 passes for the
// unfused design. f32 WMMA (v_wmma_f32_16x16x4_f32) keeps exact f32
// accumulation matching the reference.

typedef __attribute__((ext_vector_type(2))) float v2f;
typedef __attribute__((ext_vector_type(8))) float v8f;

#define MLA_B 64
#define MLA_H 32
#define D_NOPE 512
#define D_ROPE 64
#define D_TOT 576
#define PAGE_SZ 128
#define MAX_PAGES 32
#define MAX_KV 4096
#define CHUNK_PAGES 4
#define CHUNK_TOK (CHUNK_PAGES * PAGE_SZ) /* 512 tokens per split-K chunk */
#define N_CHUNKS (MAX_KV / CHUNK_TOK)     /* 8 */
#define Q_PITCH 580 /* 576 + 4: pitch%64==4 -> conflict-free ds_load_b64 */
#define P_PITCH 132 /* 128 + 4: pitch%64==4 -> conflict-free ds_load_b64 */
#define ATT_SCALE 0.0416666666666666643537020044012f /* 1/sqrt(576) */

#define SMEM_FLOATS (MLA_H * Q_PITCH + MLA_H * P_PITCH + 256 + 96)
#define SMEM_BYTES (SMEM_FLOATS * 4)

__device__ __forceinline__ v8f wmma_f32(v2f a, v2f b, v8f c) {
  // D(16x16 f32) = A(16x4 f32) * B(4x16 f32) + C
  return __builtin_amdgcn_wmma_f32_16x16x4_f32(false, a, false, b, (short)0, c,
                                               false, false);
}

// ---------------------------------------------------------------------------
// Fused flash chunk kernel. grid = (B, N_CHUNKS), block = 256 (8 waves).
// Each WG handles one sequence x one 512-token chunk and emits unnormalized
// partials: Opart[b,c,h,512], mpart[b,c,h], lpart[b,c,h].
// ---------------------------------------------------------------------------
__global__ __launch_bounds__(256) void mla_flash_chunk_kernel(
    const float* __restrict__ qn_all, const float* __restrict__ qr_all,
    const float* __restrict__ kvn_cache, const float* __restrict__ kvr_cache,
    const int* __restrict__ block_tables, const int* __restrict__ k_seqlen,
    float* __restrict__ Opart, float* __restrict__ mpart,
    float* __restrict__ lpart) {
  const int b = blockIdx.x;
  const int c = blockIdx.y;
  const int len = k_seqlen[b];
  const int tok0 = c * CHUNK_TOK;
  if (tok0 >= len) return;  // uniform: whole WG exits before any barrier
  int npg = (len - tok0 + PAGE_SZ - 1) / PAGE_SZ;
  if (npg > CHUNK_PAGES) npg = CHUNK_PAGES;

  extern __shared__ float smem[];
  float* Qs = smem;                    // [32][Q_PITCH]  (nope | rope concat)
  float* pT = Qs + MLA_H * Q_PITCH;    // [32][P_PITCH]  scores -> P~
  float* redS = pT + MLA_H * P_PITCH;  // [256] reduction scratch
  float* m_s = redS + 256;             // [32] running row max
  float* l_s = m_s + 32;               // [32] running row sum
  float* r_s = l_s + 32;               // [32] per-page rescale factor

  const int tid = threadIdx.x;
  const int wave = tid >> 5;
  const int lane = tid & 31;
  const int l15 = lane & 15;
  const int kOff = (lane >> 4) << 1;  // 0 or 2 (WMMA 32-bit A/B K split)
  const int mrow = (lane < 16) ? 0 : 8;

  // ---- Stage 0: stage Q once into LDS --------------------------------------
  for (int idx = tid; idx < MLA_H * D_TOT; idx += 256) {
    const int h = idx / D_TOT;
    const int d = idx - h * D_TOT;
    const float v = (d < D_NOPE)
                        ? qn_all[((size_t)b * MLA_H + h) * D_NOPE + d]
                        : qr_all[((size_t)b * MLA_H + h) * D_ROPE + (d - D_NOPE)];
    Qs[h * Q_PITCH + d] = v;
  }
  if (tid < MLA_H) {
    m_s[tid] = -3.0e38f;
    l_s[tid] = 0.0f;
  }
  __syncthreads();

  const int* __restrict__ bt = block_tables + b * MAX_PAGES + c * CHUNK_PAGES;

  // Stage-3 accumulators: wave -> (head tile ht3, d-quarter dq)
  const int ht3 = wave >> 2;  // 0..1
  const int dq = wave & 3;    // 0..3 -> d range [dq*128, dq*128+128)
  v8f acc[8];
#pragma unroll
  for (int dt = 0; dt < 8; ++dt)
    acc[dt] = (v8f){0.f, 0.f, 0.f, 0.f, 0.f, 0.f, 0.f, 0.f};

  for (int pg = 0; pg < npg; ++pg) {
    const int page = bt[pg];
    const float* __restrict__ vn = kvn_cache + (size_t)page * PAGE_SZ * D_NOPE;
    const float* __restrict__ vr = kvr_cache + (size_t)page * PAGE_SZ * D_ROPE;

    // ---- Stage 1: S = Q . K^T for this page (wave w = 16-token tile w) ----
    {
      const float* __restrict__ vnRow = vn + (size_t)(wave * 16 + l15) * D_NOPE;
      const float* __restrict__ vrRow = vr + (size_t)(wave * 16 + l15) * D_ROPE;
      for (int ht = 0; ht < 2; ++ht) {
        const float* __restrict__ qrow = Qs + (ht * 16 + l15) * Q_PITCH;
        v8f s = (v8f){0.f, 0.f, 0.f, 0.f, 0.f, 0.f, 0.f, 0.f};
#pragma unroll 8
        for (int k = 0; k < D_NOPE; k += 4) {
          v2f a = *(const v2f*)(qrow + k + kOff);          // LDS (ds_load_b64)
          v2f bb = *(const v2f*)(vnRow + k + kOff);        // global b64
          s = wmma_f32(a, bb, s);
        }
#pragma unroll
        for (int k = 0; k < D_ROPE; k += 4) {
          v2f a = *(const v2f*)(qrow + D_NOPE + k + kOff);
          v2f bb = *(const v2f*)(vrRow + k + kOff);
          s = wmma_f32(a, bb, s);
        }
#pragma unroll
        for (int j = 0; j < 8; ++j)
          pT[(ht * 16 + mrow + j) * P_PITCH + wave * 16 + l15] =
              s[j] * ATT_SCALE;
      }
    }
    __syncthreads();

    // ---- Stage 2: online softmax update (8 threads per head) --------------
    {
      const int h = tid >> 3;
      const int sub = tid & 7;
      const int base = tok0 + pg * PAGE_SZ;
      float* __restrict__ prow = pT + h * P_PITCH;

      float mloc = -3.0e38f;
      for (int t = sub * 16; t < sub * 16 + 16; ++t)
        if (base + t < len) mloc = fmaxf(mloc, prow[t]);
      redS[tid] = mloc;
      __syncthreads();
      if (sub == 0) {
        float pmax = redS[tid];
        for (int u = 1; u < 8; ++u) pmax = fmaxf(pmax, redS[tid + u]);
        const float mo = m_s[h];
        const float mn = fmaxf(mo, pmax);
        m_s[h] = mn;
        r_s[h] = __expf(mo - mn);  // page 0: exp(-3e38 - mn) == 0
      }
      __syncthreads();
      const float mn = m_s[h];
      float sloc = 0.0f;
      for (int t = sub * 16; t < sub * 16 + 16; ++t) {
        const float e = (base + t < len) ? __expf(prow[t] - mn) : 0.0f;
        prow[t] = e;  // P~ (unnormalized)
        sloc += e;
      }
      redS[tid] = sloc;
      __syncthreads();
      if (sub == 0) {
        float ps = 0.0f;
        for (int u = 0; u < 8; ++u) ps += redS[tid + u];
        l_s[h] = l_s[h] * r_s[h] + ps;
      }
    }
    __syncthreads();

    // ---- Stage 3: O = O * r + P~ x V (V page is L2-hot from stage 1) ------
    {
#pragma unroll
      for (int j = 0; j < 8; ++j) {
        const float r = r_s[ht3 * 16 + mrow + j];  // LDS broadcast
#pragma unroll
        for (int dt = 0; dt < 8; ++dt) acc[dt][j] *= r;
      }
      const float* __restrict__ prow = pT + (ht3 * 16 + l15) * P_PITCH;
#pragma unroll 2
      for (int kk = 0; kk < PAGE_SZ; kk += 4) {
        v2f a = *(const v2f*)(prow + kk + kOff);  // P~ pair (LDS)
        const float* __restrict__ vrow = vn + (size_t)(kk + kOff) * D_NOPE;
#pragma unroll
        for (int dt = 0; dt < 8; ++dt) {
          const int d = dq * 128 + dt * 16 + l15;
          v2f bb = {vrow[d], vrow[D_NOPE + d]};  // adjacent tokens, same d
          acc[dt] = wmma_f32(a, bb, acc[dt]);
        }
      }
    }
    __syncthreads();  // protect pT before next page's stage 1
  }

  // ---- Emit chunk partials -------------------------------------------------
  float* __restrict__ Ob = Opart + ((size_t)b * N_CHUNKS + c) * MLA_H * D_NOPE;
#pragma unroll
  for (int j = 0; j < 8; ++j) {
    const int head = ht3 * 16 + mrow + j;
#pragma unroll
    for (int dt = 0; dt < 8; ++dt) {
      const int d = dq * 128 + dt * 16 + l15;
      Ob[(size_t)head * D_NOPE + d] = acc[dt][j];
    }
  }
  if (tid < MLA_H) {
    mpart[((size_t)b * N_CHUNKS + c) * MLA_H + tid] = m_s[tid];
    lpart[((size_t)b * N_CHUNKS + c) * MLA_H + tid] = l_s[tid];
  }
}

// ---------------------------------------------------------------------------
// Combine kernel: merge <=8 chunk partials per (b, h) row.
// grid = B*H, block = 256.
// ---------------------------------------------------------------------------
__global__ __launch_bounds__(256) void mla_combine_kernel(
    const float* __restrict__ Opart, const float* __restrict__ mpart,
    const float* __restrict__ lpart, const int* __restrict__ k_seqlen,
    float* __restrict__ out) {
  const int row = blockIdx.x;  // b*H + h
  const int b = row / MLA_H;
  const int h = row - b * MLA_H;
  const int len = k_seqlen[b];
  const int nc = (len + CHUNK_TOK - 1) / CHUNK_TOK;  // 1..8

  float M = -3.0e38f;
#pragma unroll
  for (int cc = 0; cc < N_CHUNKS; ++cc)
    if (cc < nc)
      M = fmaxf(M, mpart[((size_t)b * N_CHUNKS + cc) * MLA_H + h]);

  float w[N_CHUNKS];
  float L = 0.0f;
#pragma unroll
  for (int cc = 0; cc < N_CHUNKS; ++cc) {
    float wc = 0.0f;
    if (cc < nc) {
      wc = __expf(mpart[((size_t)b * N_CHUNKS + cc) * MLA_H + h] - M);
      L += wc * lpart[((size_t)b * N_CHUNKS + cc) * MLA_H + h];
    }
    w[cc] = wc;
  }
  const float invL = 1.0f / L;

  for (int d = threadIdx.x; d < D_NOPE; d += 256) {
    float o = 0.0f;
#pragma unroll
    for (int cc = 0; cc < N_CHUNKS; ++cc)
      if (cc < nc)
        o += w[cc] *
             Opart[(((size_t)b * N_CHUNKS + cc) * MLA_H + h) * D_NOPE + d];
    out[((size_t)b * MLA_H + h) * D_NOPE + d] = o * invL;
  }
}

extern "C" void kernel_launch(void* const* d_in, const int* in_sizes, int n_in,
                              void* d_out, int out_size, void* d_ws,
                              size_t ws_size, hipStream_t stream) {
  (void)in_sizes; (void)n_in; (void)out_size; (void)ws_size;
  const float* qn = (const float*)d_in[0];
  const float* qr = (const float*)d_in[1];
  const float* kvn = (const float*)d_in[2];
  const float* kvr = (const float*)d_in[3];
  const int* bt = (const int*)d_in[4];
  /* d_in[5] = q_seqlen_list (all ones, decode) -- unused */
  const int* klen = (const int*)d_in[6];
  float* out = (float*)d_out;

  // workspace: Opart 32MiB + mpart/lpart 64KiB each
  float* Opart = (float*)d_ws;
  float* mpart = Opart + (size_t)MLA_B * N_CHUNKS * MLA_H * D_NOPE;
  float* lpart = mpart + (size_t)MLA_B * N_CHUNKS * MLA_H;

  dim3 gFlash(MLA_B, N_CHUNKS);
  mla_flash_chunk_kernel<<<gFlash, 256, SMEM_BYTES, stream>>>(
      qn, qr, kvn, kvr, bt, klen, Opart, mpart, lpart);
  mla_combine_kernel<<<MLA_B * MLA_H, 256, 0, stream>>>(Opart, mpart, lpart,
                                                        klen, out);
}